// FMFMNeuronInhib_42262478192714
// MI455X (gfx1250) — compile-verified
//
#include <hip/hip_runtime.h>
#include <stdint.h>

// ---------------------------------------------------------------------------
// FMFMNeuronInhib: T=4096 sequential scan over B=4096 independent LIF chains.
// Memory-bound (134MB in / 268MB out, ~0.4 FLOP/byte -> ~17us floor @23.3TB/s).
// Parallelism is capped at B lanes (nonlinear reset => no parallel scan), so
// the win on MI455X is deep latency hiding via ASYNC global->LDS copies
// (ASYNCcnt, up to 63 outstanding per wave, zero VGPR cost) + s_wait_asynccnt.
// Outputs are streamed with non-temporal stores (write-once, never re-read;
// don't evict the input stream from the 192MB L2).
// ---------------------------------------------------------------------------

#ifndef __has_builtin
#define __has_builtin(x) 0
#endif

#define AS_GLOBAL __attribute__((address_space(1)))
#define AS_LDS    __attribute__((address_space(3)))

// Element type of the async b64 copy: one 2xi32 (8B) per lane (clang's builtin
// signature: 'vector_size(8) int' pointers in AS1/AS3).
typedef int v2i __attribute__((vector_size(8)));

#if defined(__HIP_DEVICE_COMPILE__) && __has_builtin(__builtin_amdgcn_global_load_async_to_lds_b64)
#define HAVE_ASYNC_LDS 1
#else
#define HAVE_ASYNC_LDS 0
#endif

#if defined(__HIP_DEVICE_COMPILE__) && __has_builtin(__builtin_amdgcn_s_wait_asynccnt)
#define WAIT_ASYNCCNT(n) __builtin_amdgcn_s_wait_asynccnt(n)
#elif defined(__HIP_DEVICE_COMPILE__)
#define WAIT_ASYNCCNT(n) asm volatile("s_wait_asynccnt %0" :: "i"(n) : "memory")
#else
#define WAIT_ASYNCCNT(n)
#endif

namespace {
constexpr int   BLOCK    = 64;   // 2 wave32 per workgroup; B/BLOCK = 64 WGs
constexpr int   CHUNK    = 32;   // timesteps staged per pipeline stage
constexpr float BETA_MEM = 0.9f;
constexpr float BETA_INH = 0.6f;
constexpr float THRESH   = 1.0f;
}

__global__ __launch_bounds__(BLOCK) void snn_scan_kernel(
    const float* __restrict__ seq,      // (T, B, 2)
    const float* __restrict__ w_exc,    // (1, 2)
    const float* __restrict__ w_inh_p,  // scalar
    float* __restrict__ out,            // [spk | exc | inh | mem], each (T, B)
    int T, int B)
{
    // Double-buffered stage: [stage][t_local][lane] -> (x0, x1). 32 KB LDS.
    __shared__ float2 buf[2][CHUNK][BLOCK];

    const int tid = threadIdx.x;
    const int b   = blockIdx.x * BLOCK + tid;
    if (b >= B) return;   // grid is exact; keeps EXEC all-ones in practice

    const float w0 = w_exc[0];
    const float w1 = w_exc[1];
    const float wI = *w_inh_p;

    const size_t TB = (size_t)T * (size_t)B;
    float* __restrict__ o_spk = out;
    float* __restrict__ o_exc = out + TB;
    float* __restrict__ o_inh = out + 2 * TB;
    float* __restrict__ o_mem = out + 3 * TB;

    const size_t row = (size_t)B * 2;            // floats per timestep
    const float* src = seq + (size_t)b * 2;      // this lane's column at t=0

    float mem = 0.0f;
    float inh = 0.0f;

    // One recurrence step + 4 coalesced non-temporal output stores.
    // 32-bit element index (T*B < 2^31/4 per stream) -> SADDR+voffset stores.
    auto step = [&](float x0, float x1, unsigned t) {
        float cur_exc = fmaf(x1, w1, x0 * w0);        // x_t @ w_exc.T
        inh           = fmaf(BETA_INH, inh, x0);      // inhibition filter
        float cur_inh = wI * inh;
        float cur     = cur_exc + cur_inh;
        float reset   = (mem > THRESH) ? 1.0f : 0.0f; // spike_fn(mem - 1)
        mem           = fmaf(BETA_MEM, mem, cur) - reset * THRESH;
        float spk     = (mem > THRESH) ? 1.0f : 0.0f; // spike_fn(mem_new - 1)
        unsigned idx  = t * (unsigned)B + (unsigned)b;
        __builtin_nontemporal_store(spk,     o_spk + idx);
        __builtin_nontemporal_store(cur_exc, o_exc + idx);
        __builtin_nontemporal_store(cur_inh, o_inh + idx);
        __builtin_nontemporal_store(mem,     o_mem + idx);
    };

    // Issue one chunk of async global->LDS copies. Each instruction moves a
    // contiguous 256B row slice (32 lanes x 8B) for one timestep; CHUNK=32
    // keeps <=63 outstanding on ASYNCcnt even while double-buffered.
    auto issue = [&](int c, int s) {
        const float* g = src + (size_t)c * CHUNK * row;
#if HAVE_ASYNC_LDS
        #pragma unroll
        for (int t = 0; t < CHUNK; ++t) {
            __builtin_amdgcn_global_load_async_to_lds_b64(
                (AS_GLOBAL v2i*)(void*)const_cast<float*>(g),
                (AS_LDS    v2i*)(void*)&buf[s][t][tid],
                /*imm offset*/0, /*cpol*/0);
            g += row;
        }
#else
        // Fallback (builtin unavailable): regular loads staged through LDS.
        #pragma unroll
        for (int t = 0; t < CHUNK; ++t) {
            buf[s][t][tid] = *(const float2*)g;
            g += row;
        }
#endif
    };

    const int nfull = T / CHUNK;

    if (nfull > 0) issue(0, 0);

    for (int c = 0; c < nfull; ++c) {
        if (c + 1 < nfull) {
            issue(c + 1, (c + 1) & 1);
            // Async loads complete in order: <=CHUNK outstanding means the
            // CHUNK loads of chunk c have all landed in LDS.
            WAIT_ASYNCCNT(CHUNK);
        } else {
            WAIT_ASYNCCNT(0);
        }
        asm volatile("" ::: "memory");   // keep ds reads below the wait

        const unsigned tbase = (unsigned)(c * CHUNK);
        #pragma unroll 4
        for (int t = 0; t < CHUNK; ++t) {
            float2 v = buf[c & 1][t][tid];
            step(v.x, v.y, tbase + (unsigned)t);
        }
        asm volatile("" ::: "memory");   // finish consuming before buffer reuse
    }

    // Tail (T % CHUNK) via direct loads.
    for (int t = nfull * CHUNK; t < T; ++t) {
        const float* g = src + (size_t)t * row;
        step(g[0], g[1], (unsigned)t);
    }
}

extern "C" void kernel_launch(void* const* d_in, const int* in_sizes, int n_in,
                              void* d_out, int out_size, void* d_ws, size_t ws_size,
                              hipStream_t stream) {
    const float* spike_seq = (const float*)d_in[0];   // (T, B, 2) fp32
    const float* w_exc     = (const float*)d_in[1];   // (1, 2)   fp32
    const float* w_inh     = (const float*)d_in[2];   // scalar   fp32
    float*       out       = (float*)d_out;           // 4 x (T, B) fp32

    const int B = 4096;
    const int T = in_sizes[0] / (B * 2);              // = 4096

    dim3 grid((B + BLOCK - 1) / BLOCK);
    dim3 block(BLOCK);
    snn_scan_kernel<<<grid, block, 0, stream>>>(spike_seq, w_exc, w_inh, out, T, B);
}